// GNN_Layer2_34832184770735
// MI455X (gfx1250) — compile-verified
//
#include <hip/hip_runtime.h>

#define NN 50000      // nodes
#define NE 800000     // edges
#define DD 64
#define HH 4
#define EPSV 1e-16f

typedef float v2f __attribute__((ext_vector_type(2)));
typedef float v8f __attribute__((ext_vector_type(8)));

// D(16x16,f32) = A(16x4,f32) x B(4x16,f32) + C   — CDNA5 V_WMMA_F32_16X16X4_F32
#define WMMA4(A, B, C) \
  __builtin_amdgcn_wmma_f32_16x16x4_f32(false, (A), false, (B), (short)0, (C), false, false)

__device__ __forceinline__ float lrelu(float x, float s) { return x > 0.f ? x : s * x; }

// Weight matrices are staged in LDS pre-swizzled into WMMA B-fragment order:
// slot s = ((nt*16 + k)*32 + lane); each slot holds the two f32 B values for
// that lane: lanes 0-15 -> K = 4k,4k+1 @ col 16nt+lane ; lanes 16-31 -> K = 4k+2,4k+3.
__device__ __forceinline__ v2f frag(const float* base, int nt, int k, int lane) {
  return *(const v2f*)(base + (((nt * 16 + k) * 32 + lane) << 1));
}

__device__ void stage_weights(const float* __restrict__ W, float* dst, int coltiles, int C,
                              int tid, int nthreads) {
  int total = coltiles * 16 * 32;
  for (int s = tid; s < total; s += nthreads) {
    int lane = s & 31, k = (s >> 5) & 15, nt = s >> 9;
    int c = nt * 16 + (lane & 15);
    int r = 4 * k + 2 * (lane >> 4);
    dst[2 * s]     = W[r * C + c];
    dst[2 * s + 1] = W[(r + 1) * C + c];
  }
}

// float <-> monotone int key (for atomicMax on floats)
__device__ __forceinline__ int fkey(float f) {
  int b = __float_as_int(f);
  return b >= 0 ? b : (b ^ 0x7fffffff);
}
__device__ __forceinline__ float funkey(int k) {
  int b = k >= 0 ? k : (k ^ 0x7fffffff);
  return __int_as_float(b);
}
#define KEY_NEG_INF ((int)0x807FFFFF)

// ---------------------------------------------------------------------------
// Kernel 0: zero/seed the reduction buffers (must run every launch)
// ---------------------------------------------------------------------------
__global__ void k_init(float* __restrict__ den, int* __restrict__ maxk, float* __restrict__ agg) {
  int i = blockIdx.x * blockDim.x + threadIdx.x;
  if (i < NN * DD) agg[i] = 0.f;
  if (i < NN * HH) { den[i] = 0.f; maxk[i] = KEY_NEG_INF; }
}

// ---------------------------------------------------------------------------
// Kernel 1: per-edge pipeline — messages, attributes, attention logits.
// One wave = one 16-edge tile. All weights resident in LDS as B-fragments.
// ---------------------------------------------------------------------------
__global__ void __launch_bounds__(256) k_edge(
    const float* __restrict__ x, const int* __restrict__ ei, const float* __restrict__ ea,
    const float* __restrict__ We, const float* __restrict__ Whu, const float* __restrict__ Whw,
    const float* __restrict__ Wlin, const float* __restrict__ Ws, const float* __restrict__ Wt,
    const float* __restrict__ attw, const float* __restrict__ Wemb, const float* __restrict__ bemb,
    float* __restrict__ msgs, float* __restrict__ alpha, float* __restrict__ attr_out) {
  extern __shared__ float sm[];
  float* fWe   = sm;            // 4096 floats (64x64 as frags)
  float* fWhu  = sm + 4096;
  float* fWhw  = sm + 8192;
  float* fWlin = sm + 12288;
  float* fWs   = sm + 16384;    // 16384 floats (64x256 as frags)
  float* fWt   = sm + 32768;
  float* fWemb = sm + 49152;
  float* sbemb = sm + 65536;    // 256
  float* satt  = sm + 65792;    // 256
  float* stage = sm + 66048;    // 8 waves * 1024 (16x64 row-major tile)

  int tid = threadIdx.x;
  stage_weights(We,   fWe,   4, 64,  tid, 256);
  stage_weights(Whu,  fWhu,  4, 64,  tid, 256);
  stage_weights(Whw,  fWhw,  4, 64,  tid, 256);
  stage_weights(Wlin, fWlin, 4, 64,  tid, 256);
  stage_weights(Ws,   fWs,   16, 256, tid, 256);
  stage_weights(Wt,   fWt,   16, 256, tid, 256);
  stage_weights(Wemb, fWemb, 16, 256, tid, 256);
  if (tid < 256) { sbemb[tid] = bemb[tid]; satt[tid] = attw[tid]; }
  __syncthreads();

  int lane = tid & 31, w = tid >> 5;
  int half = lane >> 4, r16 = lane & 15;
  float* st = stage + w * 1024;
  const int NT = NE / 16;
  for (int tile = blockIdx.x * 8 + w; tile < NT; tile += gridDim.x * 8) {
    int e = tile * 16 + r16;
    int sidx = ei[e];
    int tidx = ei[NE + e];
    const float* ear = ea + (size_t)e * DD;
    const float* xsr = x + (size_t)sidx * DD;
    const float* xtr = x + (size_t)tidx * DD;

    v2f ea_a[16], xs_a[16], xt_a[16];
#pragma unroll
    for (int k = 0; k < 16; k++) {
      int c = 4 * k + 2 * half;
      ea_a[k] = *(const v2f*)(ear + c);
      xs_a[k] = *(const v2f*)(xsr + c);
      xt_a[k] = *(const v2f*)(xtr + c);
    }

    // ---- messages = leaky(ea@We + xs@Whu + xt@Whw, 0.01) -> LDS tile
#pragma unroll
    for (int nt = 0; nt < 4; nt++) {
      v8f acc = {0.f, 0.f, 0.f, 0.f, 0.f, 0.f, 0.f, 0.f};
#pragma unroll
      for (int k = 0; k < 16; k++) {
        acc = WMMA4(ea_a[k], frag(fWe,  nt, k, lane), acc);
        acc = WMMA4(xs_a[k], frag(fWhu, nt, k, lane), acc);
        acc = WMMA4(xt_a[k], frag(fWhw, nt, k, lane), acc);
      }
#pragma unroll
      for (int v = 0; v < 8; v++)
        st[(v + 8 * half) * DD + nt * 16 + r16] = lrelu(acc[v], 0.01f);
    }
    // coalesced messages -> global scratch
    {
      float* gm = msgs + (size_t)tile * 1024;
#pragma unroll
      for (int i = 0; i < 8; i++) {
        int off = i * 128 + lane * 4;
        *(float4*)(gm + off) = *(const float4*)(st + off);
      }
    }
    // ---- attributes = (ea + messages) @ Wlin
    v2f at_a[16];
#pragma unroll
    for (int k = 0; k < 16; k++) {
      int c = 4 * k + 2 * half;
      v2f m2 = *(const v2f*)(st + r16 * DD + c);
      at_a[k] = ea_a[k] + m2;
    }
#pragma unroll
    for (int nt = 0; nt < 4; nt++) {
      v8f acc = {0.f, 0.f, 0.f, 0.f, 0.f, 0.f, 0.f, 0.f};
#pragma unroll
      for (int k = 0; k < 16; k++)
        acc = WMMA4(at_a[k], frag(fWlin, nt, k, lane), acc);
#pragma unroll
      for (int v = 0; v < 8; v++)
        st[(v + 8 * half) * DD + nt * 16 + r16] = acc[v];
    }
    {
      float* ga = attr_out + (size_t)tile * 1024;
#pragma unroll
      for (int i = 0; i < 8; i++) {
        int off = i * 128 + lane * 4;
        *(float4*)(ga + off) = *(const float4*)(st + off);
      }
    }
    // reload attributes in A-fragment layout
#pragma unroll
    for (int k = 0; k < 16; k++)
      at_a[k] = *(const v2f*)(st + r16 * DD + 4 * k + 2 * half);

    // ---- alpha[e,h] = sum_d leaky(xs@Ws + xt@Wt + attr@Wemb + b, 0.2)*att
#pragma unroll
    for (int h = 0; h < HH; h++) {
      float part[8];
#pragma unroll
      for (int v = 0; v < 8; v++) part[v] = 0.f;
#pragma unroll
      for (int q = 0; q < 4; q++) {
        int ct = 4 * h + q;
        v8f acc = {0.f, 0.f, 0.f, 0.f, 0.f, 0.f, 0.f, 0.f};
#pragma unroll
        for (int k = 0; k < 16; k++) {
          acc = WMMA4(xs_a[k], frag(fWs,   ct, k, lane), acc);
          acc = WMMA4(xt_a[k], frag(fWt,   ct, k, lane), acc);
          acc = WMMA4(at_a[k], frag(fWemb, ct, k, lane), acc);
        }
        float bias = sbemb[ct * 16 + r16];
        float aw   = satt[h * 64 + q * 16 + r16];
#pragma unroll
        for (int v = 0; v < 8; v++) {
          float s = lrelu(acc[v] + bias, 0.2f);
          part[v] += s * aw;
        }
      }
      // reduce over the 16 lanes of each half (rows live in VGPR index)
#pragma unroll
      for (int m = 1; m < 16; m <<= 1)
#pragma unroll
        for (int v = 0; v < 8; v++) part[v] += __shfl_xor(part[v], m, 32);
      if (r16 == 0) {
#pragma unroll
        for (int v = 0; v < 8; v++)
          alpha[(size_t)(tile * 16 + v + 8 * half) * HH + h] = part[v];
      }
    }
  }
}

// ---------------------------------------------------------------------------
// Segment softmax + aggregation
// ---------------------------------------------------------------------------
__global__ void k_max(const float* __restrict__ alpha, const int* __restrict__ ei,
                      int* __restrict__ maxk) {
  int i = blockIdx.x * blockDim.x + threadIdx.x;
  if (i >= NE * HH) return;
  int e = i >> 2, h = i & 3;
  int t = ei[NE + e];
  atomicMax(&maxk[t * HH + h], fkey(alpha[i]));
}

__global__ void k_exp(float* __restrict__ alpha, const int* __restrict__ ei,
                      const int* __restrict__ maxk, float* __restrict__ den) {
  int i = blockIdx.x * blockDim.x + threadIdx.x;
  if (i >= NE * HH) return;
  int e = i >> 2, h = i & 3;
  int t = ei[NE + e];
  float m = funkey(maxk[t * HH + h]);
  float ex = __expf(alpha[i] - m);
  alpha[i] = ex;
  atomicAdd(&den[t * HH + h], ex);
}

__global__ void k_abar(const float* __restrict__ alpha, const int* __restrict__ ei,
                       const float* __restrict__ den, float* __restrict__ abar) {
  int e = blockIdx.x * blockDim.x + threadIdx.x;
  if (e >= NE) return;
  int t = ei[NE + e];
  float s = 0.f;
#pragma unroll
  for (int h = 0; h < HH; h++) s += alpha[e * HH + h] / (den[t * HH + h] + EPSV);
  abar[e] = 0.25f * s;
}

__global__ void k_scatter(const float* __restrict__ msgs, const float* __restrict__ abar,
                          const int* __restrict__ ei, float* __restrict__ agg) {
  long long i = (long long)blockIdx.x * blockDim.x + threadIdx.x;
  if (i >= (long long)NE * DD) return;
  int e = (int)(i >> 6), d = (int)(i & 63);
  int t = ei[NE + e];
  atomicAdd(&agg[t * DD + d], abar[e] * msgs[i]);
}

// ---------------------------------------------------------------------------
// Kernel: embeddings = x + leaky(x@W_weight + agg@W_lin_mes, 0.01)
// ---------------------------------------------------------------------------
__global__ void __launch_bounds__(256) k_node(const float* __restrict__ x,
                                              const float* __restrict__ agg,
                                              const float* __restrict__ Ww,
                                              const float* __restrict__ Wlin,
                                              float* __restrict__ out) {
  __shared__ float sw[8192];
  stage_weights(Ww,   sw,        4, 64, threadIdx.x, 256);
  stage_weights(Wlin, sw + 4096, 4, 64, threadIdx.x, 256);
  __syncthreads();
  int lane = threadIdx.x & 31, w = threadIdx.x >> 5;
  int half = lane >> 4, r16 = lane & 15;
  const int NT = NN / 16;
  for (int tile = blockIdx.x * 8 + w; tile < NT; tile += gridDim.x * 8) {
    int base = tile * 16;
    const float* xr = x + (size_t)(base + r16) * DD;
    const float* gr = agg + (size_t)(base + r16) * DD;
    v2f xa[16], ga[16];
#pragma unroll
    for (int k = 0; k < 16; k++) {
      int c = 4 * k + 2 * half;
      xa[k] = *(const v2f*)(xr + c);
      ga[k] = *(const v2f*)(gr + c);
    }
#pragma unroll
    for (int nt = 0; nt < 4; nt++) {
      v8f acc = {0.f, 0.f, 0.f, 0.f, 0.f, 0.f, 0.f, 0.f};
#pragma unroll
      for (int k = 0; k < 16; k++) {
        acc = WMMA4(xa[k], frag(sw,        nt, k, lane), acc);
        acc = WMMA4(ga[k], frag(sw + 4096, nt, k, lane), acc);
      }
#pragma unroll
      for (int v = 0; v < 8; v++) {
        int r = v + 8 * half, c = nt * 16 + r16;
        size_t idx = (size_t)(base + r) * DD + c;
        out[idx] = x[idx] + lrelu(acc[v], 0.01f);
      }
    }
  }
}

// ---------------------------------------------------------------------------
extern "C" void kernel_launch(void* const* d_in, const int* in_sizes, int n_in,
                              void* d_out, int out_size, void* d_ws, size_t ws_size,
                              hipStream_t stream) {
  const float* x    = (const float*)d_in[0];
  const int*   ei   = (const int*)d_in[1];
  const float* ea   = (const float*)d_in[2];
  const float* We   = (const float*)d_in[3];
  const float* Whu  = (const float*)d_in[4];
  const float* Whw  = (const float*)d_in[5];
  const float* Wlin = (const float*)d_in[6];
  const float* Ww   = (const float*)d_in[7];
  const float* Ws   = (const float*)d_in[8];
  const float* Wt   = (const float*)d_in[9];
  const float* attw = (const float*)d_in[10];
  const float* Wemb = (const float*)d_in[11];
  const float* bemb = (const float*)d_in[12];

  float* emb_out  = (float*)d_out;                       // [N, 64]
  float* attr_out = emb_out + (size_t)NN * DD;           // [E, 64]

  float* ws    = (float*)d_ws;
  float* msgs  = ws;                                     // E*64
  float* alpha = msgs + (size_t)NE * DD;                 // E*4
  float* abar  = alpha + (size_t)NE * HH;                // E
  int*   maxk  = (int*)(abar + NE);                      // N*4
  float* den   = (float*)(maxk + (size_t)NN * HH);       // N*4
  float* agg   = den + (size_t)NN * HH;                  // N*64

  k_init<<<(NN * DD + 255) / 256, 256, 0, stream>>>(den, maxk, agg);

  size_t shmem = (size_t)(66048 + 8 * 1024) * sizeof(float);  // ~290 KB / WGP
  k_edge<<<512, 256, shmem, stream>>>(x, ei, ea, We, Whu, Whw, Wlin, Ws, Wt, attw,
                                      Wemb, bemb, msgs, alpha, attr_out);

  k_max<<<(NE * HH + 255) / 256, 256, 0, stream>>>(alpha, ei, maxk);
  k_exp<<<(NE * HH + 255) / 256, 256, 0, stream>>>(alpha, ei, maxk, den);
  k_abar<<<(NE + 255) / 256, 256, 0, stream>>>(alpha, ei, den, abar);
  k_scatter<<<(int)(((long long)NE * DD + 255) / 256), 256, 0, stream>>>(msgs, abar, ei, agg);

  k_node<<<(NN / 16 + 7) / 8, 256, 0, stream>>>(x, agg, Ww, Wlin, emb_out);
}